// Line_block_41188736369104
// MI455X (gfx1250) — compile-verified
//
#include <hip/hip_runtime.h>
#include <hip/hip_bf16.h>
#include <math.h>

// ---------------- CDNA5 vector types for WMMA ----------------
typedef __attribute__((ext_vector_type(2))) float v2f;
typedef __attribute__((ext_vector_type(8))) float v8f;

#define BB 8
#define GG 2048
#define KK 32
#define NN 64
#define TOTAL_ELEMS (8LL * 2048LL * 32LL * 64LL)   // 33,554,432
#define NPART 512

// ---------------------------------------------------------------------------
// Kernel 1: partial sums for global std of diff = knn_x - lc_x (broadcast)
// float4 (b128) loads: this pass is half of total HBM traffic.
// ---------------------------------------------------------------------------
__global__ void k_std_partial(const float* __restrict__ lc_x,
                              const float* __restrict__ knn_x,
                              double* __restrict__ part) {
    __shared__ double s1[256];
    __shared__ double s2[256];
    const float4* knn4 = (const float4*)knn_x;
    const float4* lc4  = (const float4*)lc_x;
    const long long Q = TOTAL_ELEMS >> 2;           // 8,388,608 float4s
    long long i0 = (long long)blockIdx.x * blockDim.x + threadIdx.x;
    long long stride = (long long)gridDim.x * blockDim.x;
    double a = 0.0, b2 = 0.0;
    for (long long q = i0; q < Q; q += stride) {
        int n4 = (int)(q & 15);                     // 16 float4 per N=64 row
        long long bg = q >> 9;                      // / (K=32 * 16)
        float4 kv = knn4[q];
        float4 lv = lc4[(bg << 4) | n4];
        float d0 = kv.x - lv.x, d1 = kv.y - lv.y;
        float d2 = kv.z - lv.z, d3 = kv.w - lv.w;
        a  += (double)d0 + (double)d1 + (double)d2 + (double)d3;
        b2 += (double)d0 * d0 + (double)d1 * d1 +
              (double)d2 * d2 + (double)d3 * d3;
    }
    s1[threadIdx.x] = a;
    s2[threadIdx.x] = b2;
    __syncthreads();
    for (int o = 128; o >= 1; o >>= 1) {
        if ((int)threadIdx.x < o) {
            s1[threadIdx.x] += s1[threadIdx.x + o];
            s2[threadIdx.x] += s2[threadIdx.x + o];
        }
        __syncthreads();
    }
    if (threadIdx.x == 0) {
        part[2 * blockIdx.x + 0] = s1[0];
        part[2 * blockIdx.x + 1] = s2[0];
    }
}

// ---------------------------------------------------------------------------
// Kernel 1b: final reduce -> inv_std = 1/(std(ddof=1) + 1e-5)
// ---------------------------------------------------------------------------
__global__ void k_std_final(const double* __restrict__ part,
                            float* __restrict__ inv_std) {
    __shared__ double s1[256];
    __shared__ double s2[256];
    double a = 0.0, b2 = 0.0;
    for (int i = threadIdx.x; i < NPART; i += 256) {
        a  += part[2 * i + 0];
        b2 += part[2 * i + 1];
    }
    s1[threadIdx.x] = a;
    s2[threadIdx.x] = b2;
    __syncthreads();
    for (int o = 128; o >= 1; o >>= 1) {
        if ((int)threadIdx.x < o) {
            s1[threadIdx.x] += s1[threadIdx.x + o];
            s2[threadIdx.x] += s2[threadIdx.x + o];
        }
        __syncthreads();
    }
    if (threadIdx.x == 0) {
        double M    = (double)TOTAL_ELEMS;
        double mean = s1[0] / M;
        double var  = (s2[0] - M * mean * mean) / (M - 1.0);
        *inv_std = (float)(1.0 / (sqrt(var) + 1e-5));
    }
}

// ---------------------------------------------------------------------------
// Kernel 2: per-(b,g) fused pipeline. One wave32 per block.
//   - ASYNC tensor-path: global_load_async_to_lds_b128 stages the raw 32x64
//     knn tile straight into LDS (no VGPR staging, tracked by ASYNCcnt)
//   - (x - lc)*inv_std fused into WMMA A-fragment construction
//   - 32x68 @ 68x32 via v_wmma_f32_16x16x4_f32 (2x2 tiles, 17 k-steps)
//   - kl_pre[b][c][g] = 2 * mean_K( sin/cos(proj) ) ; lc_line -> out[b][64+c][g]
// ---------------------------------------------------------------------------
__device__ __forceinline__ void sort3_desc(float a, float b, float c,
                                           float& hi, float& mid, float& lo) {
    hi  = fmaxf(fmaxf(a, b), c);
    lo  = fminf(fminf(a, b), c);
    mid = fmaxf(fminf(fmaxf(a, b), c), fminf(a, b));
}

__launch_bounds__(32)
__global__ void k_main(const float* __restrict__ lc_x,
                       const float* __restrict__ knn_x,
                       const float* __restrict__ fourier_B,
                       const float* __restrict__ inv_std_p,
                       float* __restrict__ kl_pre,
                       float* __restrict__ out) {
    __shared__ float S[32][68];     // cols 0..63: RAW knn tile; 64..67: features
    __shared__ float Bm[68][33];    // 2pi * fourier_B (padded)
    __shared__ float lcs[64];

    const int bg   = blockIdx.x;          // b*2048 + g
    const int b    = bg >> 11;
    const int g    = bg & 2047;
    const int lane = threadIdx.x;         // 0..31 (wave32)

    // --- async DMA of the raw 32x64 tile straight into LDS (ASYNCcnt) ---
    {
        const float* ktile = knn_x + (size_t)bg * KK * NN;
        for (int t = lane; t < 512; t += 32) {      // 512 x 16B chunks
            int e = t << 2;
            int m = e >> 6, n = e & 63;
            unsigned lds_addr = (unsigned)(unsigned long long)&S[m][n]; // flat->LDS
            unsigned goff     = (unsigned)(t << 4);                     // byte offset
            asm volatile("global_load_async_to_lds_b128 %0, %1, %2 offset:0"
                         :: "v"(lds_addr), "v"(goff), "s"(ktile)
                         : "memory");
        }
    }

    const float inv_std = *inv_std_p;

    // --- stage lc row and scaled fourier_B (overlaps with async DMA) ---
    const float* lcr = lc_x + (size_t)bg * NN;
    lcs[lane]      = lcr[lane];
    lcs[lane + 32] = lcr[lane + 32];
    for (int t = lane; t < 68 * 32; t += 32) {
        Bm[t >> 5][t & 31] = 6.28318530717958647692f * fourier_B[t];
    }
    asm volatile("s_wait_asynccnt 0" ::: "memory");
    __syncthreads();

    // --- per-neighbor features (scaled on the fly): cross + dot -> cols 64..67
    float n0, n1, n2;
    sort3_desc(lcs[0], lcs[1], lcs[2], n0, n1, n2);
    {
        const int m = lane;
        float p0, p1, p2;
        float k0 = (S[m][0] - lcs[0]) * inv_std;
        float k1 = (S[m][1] - lcs[1]) * inv_std;
        float k2 = (S[m][2] - lcs[2]) * inv_std;
        sort3_desc(k0, k1, k2, p0, p1, p2);
        float dot = 0.0f;
        #pragma unroll
        for (int n = 0; n < NN; ++n) {
            float v = (S[m][n] - lcs[n]) * inv_std;
            dot = fmaf(v, lcs[n], dot);
        }
        float f0 = p1 * n2 - p2 * n1;
        float f1 = p2 * n0 - p0 * n2;
        float f2 = p0 * n1 - p1 * n0;
        S[m][64] = f0; S[m][65] = f1; S[m][66] = f2; S[m][67] = dot;
    }
    __syncthreads();

    // --- WMMA: C(32x32) = knn_line(32x68) @ Bm(68x32), fp32 16x16x4 tiles ---
    v8f acc[2][2] = {};
    const int hl   = (lane >= 16) ? 1 : 0;   // half-wave K-pair select
    const int lidx = lane & 15;
    for (int kk = 0; kk < 64; kk += 4) {     // raw columns, scale fused here
        const int kb = kk + 2 * hl;
        v2f afrag[2], bfrag[2];
        #pragma unroll
        for (int Mi = 0; Mi < 2; ++Mi) {
            const int row = Mi * 16 + lidx;
            v2f a;
            a.x = (S[row][kb]     - lcs[kb])     * inv_std;
            a.y = (S[row][kb + 1] - lcs[kb + 1]) * inv_std;
            afrag[Mi] = a;
        }
        #pragma unroll
        for (int Nj = 0; Nj < 2; ++Nj) {
            const int col = Nj * 16 + lidx;
            v2f bv; bv.x = Bm[kb][col]; bv.y = Bm[kb + 1][col];
            bfrag[Nj] = bv;
        }
        #pragma unroll
        for (int Mi = 0; Mi < 2; ++Mi)
            #pragma unroll
            for (int Nj = 0; Nj < 2; ++Nj)
                acc[Mi][Nj] = __builtin_amdgcn_wmma_f32_16x16x4_f32(
                    false, afrag[Mi], false, bfrag[Nj],
                    (short)0, acc[Mi][Nj], false, false);
    }
    {   // K tail (kk = 64): feature columns, already final values
        const int kb = 64 + 2 * hl;
        v2f afrag[2], bfrag[2];
        #pragma unroll
        for (int Mi = 0; Mi < 2; ++Mi) {
            const int row = Mi * 16 + lidx;
            v2f a; a.x = S[row][kb]; a.y = S[row][kb + 1];
            afrag[Mi] = a;
        }
        #pragma unroll
        for (int Nj = 0; Nj < 2; ++Nj) {
            const int col = Nj * 16 + lidx;
            v2f bv; bv.x = Bm[kb][col]; bv.y = Bm[kb + 1][col];
            bfrag[Nj] = bv;
        }
        #pragma unroll
        for (int Mi = 0; Mi < 2; ++Mi)
            #pragma unroll
            for (int Nj = 0; Nj < 2; ++Nj)
                acc[Mi][Nj] = __builtin_amdgcn_wmma_f32_16x16x4_f32(
                    false, afrag[Mi], false, bfrag[Nj],
                    (short)0, acc[Mi][Nj], false, false);
    }

    // --- ff = sin/cos(proj); mean over K (rows) -> kl_pre = 2*mean ---
    #pragma unroll
    for (int Nj = 0; Nj < 2; ++Nj) {
        float ssin = 0.0f, scos = 0.0f;
        #pragma unroll
        for (int Mi = 0; Mi < 2; ++Mi)
            #pragma unroll
            for (int r = 0; r < 8; ++r) {
                float v = acc[Mi][Nj][r];
                ssin += __sinf(v);
                scos += __cosf(v);
            }
        // rows 0..7 (lanes<16) + rows 8..15 (lanes>=16), same column
        ssin += __shfl_xor(ssin, 16, 32);
        scos += __shfl_xor(scos, 16, 32);
        if (lane < 16) {
            const int j = Nj * 16 + lane;
            kl_pre[((size_t)b * 64 + j)      * GG + g] = ssin * (2.0f / 32.0f);
            kl_pre[((size_t)b * 64 + 32 + j) * GG + g] = scos * (2.0f / 32.0f);
        }
    }

    // --- lc_line = lc / ||lc||2 -> out second half (b, 64+c, g) ---
    float partial = lcs[lane] * lcs[lane] + lcs[lane + 32] * lcs[lane + 32];
    #pragma unroll
    for (int o = 16; o >= 1; o >>= 1) partial += __shfl_xor(partial, o, 32);
    const float rn = 1.0f / sqrtf(partial);
    out[((size_t)b * 128 + 64 + lane)      * GG + g] = lcs[lane]      * rn;
    out[((size_t)b * 128 + 64 + lane + 32) * GG + g] = lcs[lane + 32] * rn;
}

// ---------------------------------------------------------------------------
// Kernel 3: per (b,c): kl = gelu(kl_pre / ||kl_pre||_g); out = kl - lc_line
// ---------------------------------------------------------------------------
__global__ void k_norm(const float* __restrict__ kl_pre,
                       float* __restrict__ out) {
    __shared__ float red[256];
    const int bc = blockIdx.x;            // b*64 + c
    const int b  = bc >> 6, c = bc & 63;
    const float* row = kl_pre + (size_t)bc * GG;
    float s = 0.0f;
    for (int g = threadIdx.x; g < GG; g += 256) {
        float v = row[g];
        s = fmaf(v, v, s);
    }
    red[threadIdx.x] = s;
    __syncthreads();
    for (int o = 128; o >= 1; o >>= 1) {
        if ((int)threadIdx.x < o) red[threadIdx.x] += red[threadIdx.x + o];
        __syncthreads();
    }
    const float rn = 1.0f / sqrtf(red[0]);
    for (int g = threadIdx.x; g < GG; g += 256) {
        float x  = row[g] * rn;
        float gl = 0.5f * x * (1.0f + erff(x * 0.70710678118654752f)); // exact gelu
        float ll = out[((size_t)b * 128 + 64 + c) * GG + g];
        out[((size_t)b * 128 + c) * GG + g] = gl - ll;
    }
}

// ---------------------------------------------------------------------------
extern "C" void kernel_launch(void* const* d_in, const int* in_sizes, int n_in,
                              void* d_out, int out_size, void* d_ws, size_t ws_size,
                              hipStream_t stream) {
    const float* lc_x      = (const float*)d_in[0];   // (8,2048,64)
    const float* knn_x     = (const float*)d_in[1];   // (8,2048,32,64)
    const float* fourier_B = (const float*)d_in[2];   // (68,32)
    float* out = (float*)d_out;                       // (8,128,2048)

    char* ws = (char*)d_ws;
    double* partials = (double*)ws;                   // NPART*2 doubles (8 KB)
    float*  inv_std  = (float*)(ws + 16384);          // 1 float
    float*  kl_pre   = (float*)(ws + 32768);          // 8*64*2048 floats (4 MB)

    k_std_partial<<<NPART, 256, 0, stream>>>(lc_x, knn_x, partials);
    k_std_final<<<1, 256, 0, stream>>>(partials, inv_std);
    k_main<<<BB * GG, 32, 0, stream>>>(lc_x, knn_x, fourier_B, inv_std, kl_pre, out);
    k_norm<<<BB * 64, 256, 0, stream>>>(kl_pre, out);
}